// GCNConv_15247133900890
// MI455X (gfx1250) — compile-verified
//
#include <hip/hip_runtime.h>
#include <hip/hip_bf16.h>

typedef __attribute__((ext_vector_type(2))) float v2f;
typedef __attribute__((ext_vector_type(8))) float v8f;

#define IN_F 64
#define OUT_F 64

// -------------------------------------------------------------------------
// Kernel 1: support = X @ W using V_WMMA_F32_16X16X4_F32 (exact f32 math).
// Block = 256 threads = 8 waves. Each wave computes a 16-row x 64-col tile:
// four 16x16 f32 accumulators, K marched in steps of 4 (16 WMMA issues).
// W (64x64 f32 = 16KB) staged in LDS once per block.
// -------------------------------------------------------------------------
__global__ void __launch_bounds__(256)
gcn_gemm_wmma(const float* __restrict__ X, const float* __restrict__ W,
              float* __restrict__ S, int N) {
    __shared__ float Wl[IN_F * OUT_F];

    // cooperative vectorized load of W into LDS
    for (int i = threadIdx.x * 4; i < IN_F * OUT_F; i += blockDim.x * 4) {
        *(float4*)&Wl[i] = *(const float4*)&W[i];
    }
    __syncthreads();

    const int wave = threadIdx.x >> 5;            // 0..7
    const int lane = threadIdx.x & 31;
    const int rowTile = blockIdx.x * 8 + wave;    // 16-row tile index
    const int row0 = rowTile * 16;
    if (row0 >= N) return;                        // wave-uniform exit (EXEC stays full)

    const int mlane = lane & 15;                  // M (or N) position within tile
    const int khalf = lane >> 4;                  // 0: holds K=0,1   1: holds K=2,3

    // clamp A row for safe loads on the ragged last tile (stores are guarded)
    int arow = row0 + mlane;
    if (arow >= N) arow = N - 1;
    const float* __restrict__ Arow = X + (size_t)arow * IN_F;

    v8f acc0 = {}, acc1 = {}, acc2 = {}, acc3 = {};

    for (int k0 = 0; k0 < IN_F; k0 += 4) {
        const int ka = k0 + khalf * 2;
        // A fragment: 16x4 f32 tile. lane L<16: {A[M=L][k0], A[M=L][k0+1]},
        // lanes 16-31: {A[M][k0+2], A[M][k0+3]} -- contiguous 8B load.
        v2f a = *(const v2f*)(Arow + ka);

        // B fragments from LDS: b.x = W[ka][n], b.y = W[ka+1][n]
        const float* __restrict__ Wk0 = &Wl[(size_t)ka * OUT_F];
        const float* __restrict__ Wk1 = &Wl[(size_t)(ka + 1) * OUT_F];

        v2f b0; b0.x = Wk0[ 0 + mlane]; b0.y = Wk1[ 0 + mlane];
        v2f b1; b1.x = Wk0[16 + mlane]; b1.y = Wk1[16 + mlane];
        v2f b2; b2.x = Wk0[32 + mlane]; b2.y = Wk1[32 + mlane];
        v2f b3; b3.x = Wk0[48 + mlane]; b3.y = Wk1[48 + mlane];

        acc0 = __builtin_amdgcn_wmma_f32_16x16x4_f32(false, a, false, b0, (short)0, acc0, false, false);
        acc1 = __builtin_amdgcn_wmma_f32_16x16x4_f32(false, a, false, b1, (short)0, acc1, false, false);
        acc2 = __builtin_amdgcn_wmma_f32_16x16x4_f32(false, a, false, b2, (short)0, acc2, false, false);
        acc3 = __builtin_amdgcn_wmma_f32_16x16x4_f32(false, a, false, b3, (short)0, acc3, false, false);
    }

    // C/D layout: VGPR i holds M=i (lanes 0-15) and M=i+8 (lanes 16-31), N=lane&15.
    const int mofs = khalf * 8;
#pragma unroll
    for (int i = 0; i < 8; ++i) {
        const int r = row0 + i + mofs;
        if (r < N) {
            float* __restrict__ Srow = S + (size_t)r * OUT_F + mlane;
            Srow[ 0] = acc0[i];
            Srow[16] = acc1[i];
            Srow[32] = acc2[i];
            Srow[48] = acc3[i];
        }
    }
}

// -------------------------------------------------------------------------
// Kernel 2: out[n][f] = bias[f]   (also provides required zero-state init)
// -------------------------------------------------------------------------
__global__ void __launch_bounds__(256)
gcn_init_bias(float* __restrict__ out, const float* __restrict__ bias, int total) {
    int i = blockIdx.x * blockDim.x + threadIdx.x;
    if (i < total) out[i] = bias[i & (OUT_F - 1)];
}

// -------------------------------------------------------------------------
// Kernel 3: edge scatter. One wave per edge; each lane owns a float2 of
// features -> coalesced 256B gather + 2 global_atomic_add_f32 per lane.
// -------------------------------------------------------------------------
__global__ void __launch_bounds__(256)
gcn_spmm_scatter(const float* __restrict__ S, const int* __restrict__ erow,
                 const int* __restrict__ ecol, const float* __restrict__ eval,
                 float* __restrict__ out, int E) {
    const int gwave = (int)((blockIdx.x * blockDim.x + threadIdx.x) >> 5);
    const int lane = threadIdx.x & 31;
    if (gwave >= E) return;

    const int r = erow[gwave];
    const int c = ecol[gwave];
    const float v = eval[gwave];

    const float2 sv = *(const float2*)(S + (size_t)c * OUT_F + lane * 2);
    float* __restrict__ op = out + (size_t)r * OUT_F + lane * 2;
    atomicAdd(op + 0, sv.x * v);
    atomicAdd(op + 1, sv.y * v);
}

extern "C" void kernel_launch(void* const* d_in, const int* in_sizes, int n_in,
                              void* d_out, int out_size, void* d_ws, size_t ws_size,
                              hipStream_t stream) {
    const float* x        = (const float*)d_in[0];   // [N, 64]
    const int*   edge_row = (const int*)d_in[1];     // [E]
    const int*   edge_col = (const int*)d_in[2];     // [E]
    const float* edge_val = (const float*)d_in[3];   // [E]
    const float* weight   = (const float*)d_in[4];   // [64, 64]
    const float* bias     = (const float*)d_in[5];   // [64]
    float* out = (float*)d_out;                      // [N, 64]

    const int N = in_sizes[0] / IN_F;
    const int E = in_sizes[1];

    float* support = (float*)d_ws;                   // [N, 64] f32 scratch

    // 1) support = x @ W  (WMMA f32)
    const int rowTiles = (N + 15) / 16;
    const int gemmBlocks = (rowTiles + 7) / 8;       // 8 waves (tiles) per block
    gcn_gemm_wmma<<<gemmBlocks, 256, 0, stream>>>(x, weight, support, N);

    // 2) out = broadcast(bias)
    const int total = N * OUT_F;
    gcn_init_bias<<<(total + 255) / 256, 256, 0, stream>>>(out, bias, total);

    // 3) out[r] += v * support[c]  (one wave per edge)
    const int edgeBlocks = (E + 7) / 8;              // 8 edges (waves) per block
    gcn_spmm_scatter<<<edgeBlocks, 256, 0, stream>>>(support, edge_row, edge_col,
                                                     edge_val, out, E);
}